// Angles2Backbone_18210661335439
// MI455X (gfx1250) — compile-verified
//
#include <hip/hip_runtime.h>
#include <math.h>

// ---------------------------------------------------------------------------
// Angles2Backbone on MI455X (gfx1250, wave32).
//   input : (B=128, 3, L=2048) float32 dihedral angles
//   output: (B, 3*3*L)         float32 backbone coords
//
// 1 workgroup (512 threads = 16 waves) per batch chain of 6144 affine 4x4
// transforms. Per-thread serial product of a 12-matrix segment, then a
// block-level Hillis-Steele prefix-product of the 512 segment matrices using
// V_WMMA_F32_16X16X4_F32 (block-diagonal packing: 4 independent 4x4 products
// per WMMA, exact since K=4), then pass-2 applies the exclusive prefix and
// writes masked coords. Roofline: ~12.5 MB total traffic -> ~0.5us floor at
// 23.3 TB/s; this kernel is latency-bound, so the design maximizes resident
// waves (low VGPR use, no sincos caching) and pipelines DS->WMMA per step.
// ---------------------------------------------------------------------------

#define WG_THREADS 512
#define L_RES      2048
#define N_ATOMS    (3 * L_RES)            // 6144 per batch
#define APT        (N_ATOMS / WG_THREADS) // 12 atoms per thread (divisible by 3)
#define OUT_STRIDE (3 * N_ATOMS)          // 18432 floats per batch

typedef __attribute__((ext_vector_type(2))) float v2f;
typedef __attribute__((ext_vector_type(8))) float v8f;

// P (3x4 affine, row-major) <- P @ M(s,c,ck,sk,R); exploits structural zeros.
__device__ __forceinline__ void compose(float* __restrict__ P,
                                        float s, float c,
                                        float ck, float sk, float R) {
  // M row0 = [ c,  s*ck,  s*sk, R*c]
  // M row1 = [ s, -c*ck, -c*sk, R*s]
  // M row2 = [ 0,    sk,   -ck,   0]
  const float m01 = s * ck, m02 = s * sk;
  const float m11 = -c * ck, m12 = -c * sk;
#pragma unroll
  for (int r = 0; r < 3; ++r) {
    const float p0 = P[r * 4 + 0], p1 = P[r * 4 + 1];
    const float p2 = P[r * 4 + 2], p3 = P[r * 4 + 3];
    const float n0 = p0 * c + p1 * s;
    P[r * 4 + 0] = n0;
    P[r * 4 + 1] = p0 * m01 + p1 * m11 + p2 * sk;
    P[r * 4 + 2] = p0 * m02 + p1 * m12 - p2 * ck;
    P[r * 4 + 3] = n0 * R + p3;
  }
}

__global__ __launch_bounds__(WG_THREADS)
void a2b_scan_kernel(const float* __restrict__ input,
                     const int* __restrict__ angles_length,
                     float* __restrict__ out) {
  __shared__ float angS[3 * L_RES];        // 24 KB staged angles (live both passes)
  __shared__ float sbuf[WG_THREADS * 16];  // 32 KB scan buffer (single, read-then-write)
  __shared__ float identS[16];             // 4x4 identity

  const int tid = threadIdx.x;
  const int b   = blockIdx.x;

  // ---- stage angles coalesced into LDS -----------------------------------
  const float* ang = input + (size_t)b * 3 * L_RES;
#pragma unroll 4
  for (int idx = tid; idx < 3 * L_RES; idx += WG_THREADS) angS[idx] = ang[idx];
  if (tid < 16) identS[tid] = ((tid & 3) == (tid >> 2)) ? 1.f : 0.f;
  __syncthreads();

  // ---- per-type constants -------------------------------------------------
  const float kap[3] = {3.14159f - 1.9391f, 3.14159f - 2.061f, 3.14159f - 2.1186f};
  const float RR[3]  = {1.46f, 1.525f, 1.33f};
  float CK[3], SK[3];
#pragma unroll
  for (int u = 0; u < 3; ++u) { CK[u] = __cosf(kap[u]); SK[u] = __sinf(kap[u]); }

  // ---- pass 1: local segment product (12 compositions) -------------------
  // atom i = tid*12 + k uses angle j = i-1; type = (k+2)%3 (k>=1),
  // residue = 4*tid + (k-1)/3.  k==0 uses type 2, residue 4*tid-1 (tid>0).
  float P[12] = {1, 0, 0, 0, 0, 1, 0, 0, 0, 0, 1, 0};
#pragma unroll
  for (int k = 0; k < APT; ++k) {
    int type, r;
    if (k == 0) {
      if (tid == 0) continue;              // M_0 = I
      type = 2; r = 4 * tid - 1;
    } else {
      type = (k + 2) % 3; r = 4 * tid + (k - 1) / 3;
    }
    float s, c;
    __sincosf(angS[type * L_RES + r], &s, &c);
    compose(P, s, c, CK[type], SK[type], RR[type]);
  }

  // store full 4x4 segment product into the scan buffer
  {
    float* o = sbuf + tid * 16;
#pragma unroll
    for (int q = 0; q < 12; ++q) o[q] = P[q];
    o[12] = 0.f; o[13] = 0.f; o[14] = 0.f; o[15] = 1.f;
  }
  __syncthreads();

  // ---- block-level Hillis-Steele scan via V_WMMA_F32_16X16X4_F32 ---------
  // Per step: every wave gathers operands for 8 quads (batched DS loads),
  // barrier (all reads done), 8 back-to-back WMMAs + diagonal-block scatter
  // into the same buffer, barrier. Entries with t<d use the LDS identity as
  // the left operand, so all 512 entries are produced uniformly and EXEC is
  // all-ones at every WMMA.
  const int lane = tid & 31;
  const int m    = lane & 15;
  const int hi   = lane >> 4;          // 0: K/M rows 0..7 half, 1: upper half
  const int jblk = m >> 2;             // which of the 4 matrices this lane feeds
  const int rc   = m & 3;              // row (for A) / col (for B) within block
  const bool writer = hi ? (m >= 8) : (m < 8);
  const int jj = jblk & 1;
  const int wbase = (tid >> 5) * 32;   // first scan index owned by this wave

  for (int d = 1; d < WG_THREADS; d <<= 1) {
    v2f avs[8], bvs[8];
#pragma unroll
    for (int q = 0; q < 8; ++q) {
      const int tj = wbase + q * 4 + jblk;
      const float* Lb = (tj >= d) ? (sbuf + (tj - d) * 16) : identS; // S[t-d] or I
      const float* Rb = sbuf + tj * 16;                              // S[t]
      // A 16x4 f32: lane<16 -> K={0,1}; lane>=16 -> K={2,3}
      avs[q].x = Lb[rc * 4 + 2 * hi];
      avs[q].y = Lb[rc * 4 + 2 * hi + 1];
      // B 4x16 f32 (layout symmetric to A): VGPR k -> K=k / K=k+2 by lane half
      bvs[q].x = Rb[(2 * hi) * 4 + rc];
      bvs[q].y = Rb[(2 * hi + 1) * 4 + rc];
    }
    __syncthreads();   // all reads of sbuf complete before any write
#pragma unroll
    for (int q = 0; q < 8; ++q) {
      v8f cz = {};
      v8f dv = __builtin_amdgcn_wmma_f32_16x16x4_f32(
          false, avs[q], false, bvs[q], (short)0, cz, false, false);
      // extract diagonal 4x4 block (rows 4j..4j+3, cols 4j..4j+3)
      const float w0 = jj ? dv[4] : dv[0];
      const float w1 = jj ? dv[5] : dv[1];
      const float w2 = jj ? dv[6] : dv[2];
      const float w3 = jj ? dv[7] : dv[3];
      if (writer) {
        float* ob = sbuf + (wbase + q * 4 + jblk) * 16;
        ob[0 + rc]  = w0;
        ob[4 + rc]  = w1;
        ob[8 + rc]  = w2;
        ob[12 + rc] = w3;
      }
    }
    __syncthreads();
  }
  // 9 steps -> sbuf now holds the inclusive scan S[t].

  // ---- pass 2: apply exclusive prefix, recompute segment, emit coords ----
  float Q[12];
  if (tid == 0) {
    Q[0] = 1; Q[1] = 0; Q[2]  = 0; Q[3]  = 0;
    Q[4] = 0; Q[5] = 1; Q[6]  = 0; Q[7]  = 0;
    Q[8] = 0; Q[9] = 0; Q[10] = 1; Q[11] = 0;
  } else {
    const float* e = sbuf + (tid - 1) * 16;
#pragma unroll
    for (int q = 0; q < 12; ++q) Q[q] = e[q];
  }

  const int len3 = 3 * angles_length[b];
  float* outb = out + (size_t)b * OUT_STRIDE;
#pragma unroll
  for (int k = 0; k < APT; ++k) {
    if (!(k == 0 && tid == 0)) {
      int type, r;
      if (k == 0) { type = 2; r = 4 * tid - 1; }
      else        { type = (k + 2) % 3; r = 4 * tid + (k - 1) / 3; }
      float s, c;
      __sincosf(angS[type * L_RES + r], &s, &c);
      compose(Q, s, c, CK[type], SK[type], RR[type]);
    }
    const int i = tid * APT + k;
    const bool valid = i < len3;
    outb[i * 3 + 0] = valid ? Q[3]  : 0.f;
    outb[i * 3 + 1] = valid ? Q[7]  : 0.f;
    outb[i * 3 + 2] = valid ? Q[11] : 0.f;
  }
}

extern "C" void kernel_launch(void* const* d_in, const int* in_sizes, int n_in,
                              void* d_out, int out_size, void* d_ws, size_t ws_size,
                              hipStream_t stream) {
  (void)n_in; (void)out_size; (void)d_ws; (void)ws_size;
  const float* input = (const float*)d_in[0];
  const int* alen    = (const int*)d_in[1];
  float* out         = (float*)d_out;
  const int B        = in_sizes[1];  // angles_length has B elements
  a2b_scan_kernel<<<dim3(B), dim3(WG_THREADS), 0, stream>>>(input, alen, out);
}